// Blur_by_Kernel_704374637066
// MI455X (gfx1250) — compile-verified
//
#include <hip/hip_runtime.h>

// Per-batch 21x21 depthwise blur (reflect pad) on gfx1250 via V_WMMA_F32_16X16X4_F32.
// Band-matrix formulation: out[r,n] = sum_t S[r,t] * W_i[t,n], W_i[t,n] = ker[i,t-n].
// Each wave computes a 2x2 grid of 16x16 tiles; the band operand B (kernel row)
// is shared by all four WMMAs -> ~1.6 issue slots per WMMA, matrix-pipe bound.

typedef __attribute__((ext_vector_type(2))) float v2f;
typedef __attribute__((ext_vector_type(8))) float v8f;

#define HH   768
#define WW   768
#define KK   21
#define PP   10          // KK/2
#define TS   64          // 64x64 outputs per workgroup (4 waves x 2x2 sub-tiles of 16)
#define HSW  84          // halo rows/cols = TS + 2*PP
#define KROW 64          // zero-padded kernel row stride in LDS (21 zeros | 21 taps | 22 zeros)

__global__ void __launch_bounds__(128)
blur21_wmma_f32(const float* __restrict__ in, const float* __restrict__ ker,
                float* __restrict__ out)
{
    __shared__ float sIn[HSW * HSW];  // 84*84*4 = 28224 B, stride 84 -> conflict-free A reads
    __shared__ float sK[KK * KROW];   //  5376 B, zero-padded kernel rows

    const int tid = threadIdx.x;
    const int bx  = blockIdx.x;               // tile col (64-wide)
    const int by  = blockIdx.y;               // tile row (64-tall)
    const int img = blockIdx.z;               // b*3 + ch
    const int bat = img / 3;

    const float* __restrict__ imgIn  = in  + (size_t)img * (HH * WW);
    float*       __restrict__ imgOut = out + (size_t)img * (HH * WW);
    const float* __restrict__ kerB   = ker + bat * (KK * KK);

    const int gx0 = bx * TS;
    const int gy0 = by * TS;

    // ---- stage 84x84 halo window into LDS with reflect padding ----
    for (int idx = tid; idx < HSW * HSW; idx += 128) {
        int r  = idx / HSW;
        int c  = idx - r * HSW;
        int gy = gy0 + r - PP;
        int gx = gx0 + c - PP;
        gy = (gy < 0)   ? -gy             : gy;  // reflect (edge not repeated)
        gy = (gy >= HH) ? (2*HH - 2 - gy) : gy;
        gx = (gx < 0)   ? -gx             : gx;
        gx = (gx >= WW) ? (2*WW - 2 - gx) : gx;
        sIn[idx] = imgIn[gy * WW + gx];
    }

    // ---- stage kernel rows into LDS, zero-padded to width 64 ----
    // sK[i*64 + 21 + j] = ker[i][j] for j in [0,21); zeros elsewhere -> the
    // band gathers below never need a bounds check.
    for (int idx = tid; idx < KK * KROW; idx += 128) {
        int i = idx / KROW;
        int t = idx - i * KROW;
        float v = 0.0f;
        if (t >= KK && t < 2 * KK) v = kerB[i * KK + (t - KK)];
        sK[idx] = v;
    }

    __syncthreads();

    // ---- WMMA: 21 kernel rows x 9 K-chunks x 2x2 sub-tiles per wave ----
    const int wv   = tid >> 5;            // wave 0..3
    const int lane = tid & 31;
    const int lm   = lane & 15;           // A row M / B col N / D col N
    const int g    = lane >> 4;           // half-wave selector
    const int qr   = (wv >> 1) * 16;      // row-tile offsets: qr, qr+32
    const int qc   = (wv & 1)  * 16;      // col-tile offsets: qc, qc+32

    // Per-lane bases; bumped once per kernel row so chunk loads use immediates.
    //   A[M=lm][K=4c+2g+v] tile(R,C) = sIn[(qr+32R+lm+i)*HSW + qc+32C + 4c + 2g + v]
    //   B[K=4c+2g+v][N=lm]           = sK[i*KROW + KK + 4c + 2g + v - lm]
    const float* ap0 = &sIn[(qr + lm) * HSW + qc + 2 * g];        // row-tile 0
    const float* ap1 = ap0 + 32 * HSW;                            // row-tile 1
    const float* kp  = &sK[KK + 2 * g - lm];   // min index: 21 - 15 = 6 >= 0

    v8f acc00 = {};   // (row 0, col 0)
    v8f acc01 = {};   // (row 0, col 1)
    v8f acc10 = {};   // (row 1, col 0)
    v8f acc11 = {};   // (row 1, col 1)
    for (int i = 0; i < KK; ++i) {
        #pragma unroll
        for (int c4 = 0; c4 < 9; ++c4) {
            v2f a00, a01, a10, a11, bmat;
            a00.x  = ap0[4 * c4];          // all pairs 8B-aligned
            a00.y  = ap0[4 * c4 + 1];
            a01.x  = ap0[4 * c4 + 32];
            a01.y  = ap0[4 * c4 + 33];
            a10.x  = ap1[4 * c4];
            a10.y  = ap1[4 * c4 + 1];
            a11.x  = ap1[4 * c4 + 32];
            a11.y  = ap1[4 * c4 + 33];
            bmat.x = kp[4 * c4];           // band operand shared by all 4 WMMAs
            bmat.y = kp[4 * c4 + 1];
            acc00 = __builtin_amdgcn_wmma_f32_16x16x4_f32(
                        false, a00, false, bmat, (short)0, acc00, false, false);
            acc01 = __builtin_amdgcn_wmma_f32_16x16x4_f32(
                        false, a01, false, bmat, (short)0, acc01, false, false);
            acc10 = __builtin_amdgcn_wmma_f32_16x16x4_f32(
                        false, a10, false, bmat, (short)0, acc10, false, false);
            acc11 = __builtin_amdgcn_wmma_f32_16x16x4_f32(
                        false, a11, false, bmat, (short)0, acc11, false, false);
        }
        ap0 += HSW;   // next input row
        ap1 += HSW;
        kp  += KROW;  // next kernel row
    }

    // ---- store D: VGPR v -> M = v + 8*g, N = lm ----
    #pragma unroll
    for (int v = 0; v < 8; ++v) {
        const int gy = gy0 + qr + v + 8 * g;
        const int gx = gx0 + qc + lm;
        imgOut[(gy)      * WW + gx]      = acc00[v];
        imgOut[(gy)      * WW + gx + 32] = acc01[v];
        imgOut[(gy + 32) * WW + gx]      = acc10[v];
        imgOut[(gy + 32) * WW + gx + 32] = acc11[v];
    }
}

extern "C" void kernel_launch(void* const* d_in, const int* in_sizes, int n_in,
                              void* d_out, int out_size, void* d_ws, size_t ws_size,
                              hipStream_t stream) {
    (void)in_sizes; (void)n_in; (void)out_size; (void)d_ws; (void)ws_size;
    const float* in  = (const float*)d_in[0];   // (16,3,768,768) f32
    const float* ker = (const float*)d_in[1];   // (16,21,21) f32
    float* out = (float*)d_out;                 // (16,3,768,768) f32

    dim3 grid(WW / TS, HH / TS, 16 * 3);        // 12 x 12 x 48 workgroups
    dim3 block(128, 1, 1);                      // 4 waves (wave32)
    blur21_wmma_f32<<<grid, block, 0, stream>>>(in, ker, out);
}